// NNK_Means_28046136442929
// MI455X (gfx1250) — compile-verified
//
#include <hip/hip_runtime.h>
#include <hip/hip_bf16.h>
#include <stdint.h>

// NNK_Means forward core for MI455X (gfx1250, wave32, WMMA).
//
// Pipeline (all f32 to preserve reference numerics through top-k):
//   1) k_normalize_dict : dict_norm = l2_normalize(dictionary_atoms)      [16 MB]
//   2) k_sims_gemm      : sims = batch @ dict_norm^T via V_WMMA_F32_16X16X4_F32,
//                         double-buffered GLOBAL_LOAD_ASYNC_TO_LDS_B128 staging
//   3) k_topk           : per-row top-50 (iterative argmax, jax tie-break)
//   4) k_nnk_solve      : per-row Gram via WMMA, power-iteration lambda_max
//                         (PSD => max|eig| = lambda_max), 100 ISTA steps,
//                         L1 normalize, gather-interpolation -> d_out

typedef __attribute__((ext_vector_type(2))) float v2f;
typedef __attribute__((ext_vector_type(8))) float v8f;

#define BATCH 4096
#define DIM   512
#define ATOMS 8192
#define KSEL  50
#define KPAD  64
#define ISTA_ITERS  100
#define POWER_ITERS 50

__device__ __forceinline__ v8f wmma_f32_16x16x4(v2f a, v2f b, v8f c) {
    // 8 args: (neg_a, A, neg_b, B, c_mod, C, reuse_a, reuse_b)
    return __builtin_amdgcn_wmma_f32_16x16x4_f32(
        false, a, false, b, (short)0, c, false, false);
}

// Async copy 16 bytes global -> LDS (ASYNCcnt-tracked, no VGPR data path).
// GVS addressing: mem = SADDR(64b SGPR) + VADDR(32b byte offset). VDST = LDS
// byte address (low 32 bits of the generic shared pointer).
__device__ __forceinline__ void async_ld_b128(uint32_t lds_addr, uint32_t goff,
                                              const float* __restrict__ base) {
    asm volatile("global_load_async_to_lds_b128 %0, %1, %2"
                 :: "v"(lds_addr), "v"(goff), "s"(base) : "memory");
}
__device__ __forceinline__ void wait_asynccnt0() {
    asm volatile("s_wait_asynccnt 0x0" ::: "memory");
}

// ---------------------------------------------------------------------------
// Kernel 1: row L2-normalize the dictionary (8192 rows x 512)
// ---------------------------------------------------------------------------
__global__ __launch_bounds__(128)
void k_normalize_dict(const float* __restrict__ dict, float* __restrict__ dictn) {
    const int row = blockIdx.x;
    const int tid = threadIdx.x;            // 128 threads, 1 float4 each
    __shared__ float red[128];

    const float4 v = ((const float4*)(dict + (size_t)row * DIM))[tid];
    red[tid] = v.x * v.x + v.y * v.y + v.z * v.z + v.w * v.w;
    __syncthreads();
    for (int off = 64; off > 0; off >>= 1) {
        if (tid < off) red[tid] += red[tid + off];
        __syncthreads();
    }
    const float inv = 1.0f / fmaxf(sqrtf(red[0]), 1e-12f);
    float4 o;
    o.x = v.x * inv; o.y = v.y * inv; o.z = v.z * inv; o.w = v.w * inv;
    ((float4*)(dictn + (size_t)row * DIM))[tid] = o;
}

// ---------------------------------------------------------------------------
// Kernel 2: sims(4096x8192) = batch(4096x512) @ dict_norm^T via f32 WMMA.
// Block tile 128x64, 128 threads = 4 waves; wave w owns rows [32w,32w+32):
// 8 accumulators (2 M-tiles x 4 N-tiles) -> each B fragment reused twice.
// K staged 32/step into double-buffered LDS by async B128 loads.
// LDS row stride 36 floats = 144 B: float4 dests stay 16B-aligned and
// column reads (36*l mod 64) are bank-conflict-free.
// ---------------------------------------------------------------------------
#define BM 128
#define BN 64
#define BK 32
#define LST 36

__global__ __launch_bounds__(128)
void k_sims_gemm(const float* __restrict__ batch, const float* __restrict__ dictn,
                 float* __restrict__ sims) {
    __shared__ __align__(16) float As[2][BM][LST];  // 36.0 KB
    __shared__ __align__(16) float Bs[2][BN][LST];  // 18.0 KB

    const int tid  = threadIdx.x;
    const int lane = tid & 31;
    const int wave = tid >> 5;          // 0..3
    const int half = lane >> 4;         // 0/1 -> K pair select (ISA A layout)
    const int l    = lane & 15;
    const int m0 = blockIdx.y * BM;
    const int n0 = blockIdx.x * BN;

    const uint32_t asb = (uint32_t)(uintptr_t)&As[0][0][0];
    const uint32_t bsb = (uint32_t)(uintptr_t)&Bs[0][0][0];

    auto stage = [&](int kc, int bsel) {
        // A: 128x32 floats = 1024 float4 slots, 8 per thread
        #pragma unroll
        for (int q = 0; q < 8; ++q) {
            const int lin = tid + 128 * q;
            const int row = lin >> 3;            // 0..127
            const int c4  = (lin & 7) << 2;      // 0..28
            const uint32_t lds = asb +
                (uint32_t)(((bsel * BM + row) * LST + c4) * 4);
            const uint32_t gof =
                (uint32_t)(((m0 + row) * DIM + kc + c4) * 4);
            async_ld_b128(lds, gof, batch);
        }
        // B: 64x32 floats = 512 float4 slots, 4 per thread
        #pragma unroll
        for (int q = 0; q < 4; ++q) {
            const int lin = tid + 128 * q;
            const int row = lin >> 3;            // 0..63
            const int c4  = (lin & 7) << 2;
            const uint32_t lds = bsb +
                (uint32_t)(((bsel * BN + row) * LST + c4) * 4);
            const uint32_t gof =
                (uint32_t)(((n0 + row) * DIM + kc + c4) * 4);
            async_ld_b128(lds, gof, dictn);
        }
    };

    v8f acc0[4] = {};
    v8f acc1[4] = {};

    stage(0, 0);                                  // prologue
    for (int c = 0; c < DIM / BK; ++c) {
        const int bsel = c & 1;
        wait_asynccnt0();                         // chunk c landed in LDS
        __syncthreads();                          // all waves' chunks visible;
                                                  // compute on bsel^1 finished
        if (c + 1 < DIM / BK)
            stage((c + 1) * BK, bsel ^ 1);        // overlap next chunk's DMA

        #pragma unroll
        for (int kk = 0; kk < BK; kk += 4) {
            // A 16x4: lanes 0-15 hold K=kk,kk+1; lanes 16-31 hold K=kk+2,kk+3
            v2f a0, a1;
            a0[0] = As[bsel][wave * 32 + l][kk + 2 * half];
            a0[1] = As[bsel][wave * 32 + l][kk + 2 * half + 1];
            a1[0] = As[bsel][wave * 32 + 16 + l][kk + 2 * half];
            a1[1] = As[bsel][wave * 32 + 16 + l][kk + 2 * half + 1];
            #pragma unroll
            for (int nt = 0; nt < 4; ++nt) {
                v2f b;   // B[k][n] = dict_norm[n][k] (mirrored lane layout)
                b[0] = Bs[bsel][nt * 16 + l][kk + 2 * half];
                b[1] = Bs[bsel][nt * 16 + l][kk + 2 * half + 1];
                acc0[nt] = wmma_f32_16x16x4(a0, b, acc0[nt]);
                acc1[nt] = wmma_f32_16x16x4(a1, b, acc1[nt]);
            }
        }
        __syncthreads();                          // done reading bsel before
                                                  // its DMA overwrite next+1
    }

    // C/D layout: VGPR r -> M=r (lanes 0-15) / M=r+8 (lanes 16-31), N=lane&15
    #pragma unroll
    for (int nt = 0; nt < 4; ++nt) {
        const int col = n0 + nt * 16 + l;
        #pragma unroll
        for (int r = 0; r < 8; ++r) {
            const int mrow = wave * 32 + (half ? r + 8 : r);
            sims[(size_t)(m0 + mrow) * ATOMS + col]      = acc0[nt][r];
            sims[(size_t)(m0 + mrow + 16) * ATOMS + col] = acc1[nt][r];
        }
    }
}

// ---------------------------------------------------------------------------
// Kernel 3: top-50 per row (descending, lower index on ties, like lax.top_k).
// Row cached in LDS (32 KB), 50 rounds of block-wide argmax + mask.
// ---------------------------------------------------------------------------
__global__ __launch_bounds__(256)
void k_topk(const float* __restrict__ sims,
            float* __restrict__ sub_sims, int* __restrict__ sub_idx) {
    const int row = blockIdx.x;
    const int tid = threadIdx.x;
    __shared__ float buf[ATOMS];
    __shared__ float rv[256];
    __shared__ int   ri[256];

    const float4* src  = (const float4*)(sims + (size_t)row * ATOMS);
    float4*       dst4 = (float4*)buf;
    #pragma unroll
    for (int q = 0; q < (ATOMS / 4) / 256; ++q)   // 8 float4 per thread
        dst4[tid + 256 * q] = src[tid + 256 * q];
    __syncthreads();

    for (int it = 0; it < KSEL; ++it) {
        float bv = -3.4e38f; int bi = 0;
        for (int j = tid; j < ATOMS; j += 256) {
            const float v = buf[j];
            if (v > bv) { bv = v; bi = j; }       // ascending scan => lowest idx on tie
        }
        rv[tid] = bv; ri[tid] = bi;
        __syncthreads();
        for (int s = 128; s > 0; s >>= 1) {
            if (tid < s) {
                const float ov = rv[tid + s]; const int oi = ri[tid + s];
                if (ov > rv[tid] || (ov == rv[tid] && oi < ri[tid])) {
                    rv[tid] = ov; ri[tid] = oi;
                }
            }
            __syncthreads();
        }
        if (tid == 0) {
            sub_sims[(size_t)row * KPAD + it] = rv[0];
            sub_idx [(size_t)row * KPAD + it] = ri[0];
            buf[ri[0]] = -3.4e38f;
        }
        __syncthreads();
    }
    if (tid >= KSEL && tid < KPAD) {              // zero padding rows 50..63
        sub_sims[(size_t)row * KPAD + tid] = 0.0f;
        sub_idx [(size_t)row * KPAD + tid] = 0;
    }
}

// ---------------------------------------------------------------------------
// Kernel 4: per-row solve. Gram(64x64, zero-padded) via WMMA, power iteration
// for lambda_max (PSD => max|eig|), 100 ISTA steps, L1 norm, interpolation.
// One workgroup (128 threads = 4 waves) per batch row.
// ---------------------------------------------------------------------------
__global__ __launch_bounds__(128)
void k_nnk_solve(const float* __restrict__ dictn, const float* __restrict__ atoms,
                 const float* __restrict__ sub_sims, const int* __restrict__ sub_idx,
                 float* __restrict__ out) {
    const int row  = blockIdx.x;
    const int tid  = threadIdx.x;
    const int lane = tid & 31;
    const int wave = tid >> 5;
    const int half = lane >> 4;
    const int l    = lane & 15;

    __shared__ float Sc[KPAD][BK + 1];       // support K-chunk (64x32)
    __shared__ float G[KPAD][KPAD + 1];      // Gram, padded stride
    __shared__ float xsh[KPAD], bsh[KPAD], vsh[KPAD], tsh[KPAD];
    __shared__ float red[128];
    __shared__ int   idxsh[KPAD];
    __shared__ float etash;

    if (tid < KPAD) {
        idxsh[tid] = sub_idx[(size_t)row * KPAD + tid];
        bsh[tid]   = (tid < KSEL) ? sub_sims[(size_t)row * KPAD + tid] : 0.0f;
    }
    __syncthreads();

    // ---- Gram = S * S^T, S = gathered dict_norm rows (zero rows >= 50) ----
    v8f acc[4] = {};
    const int r2  = tid >> 1;                // 0..63
    const int h16 = (tid & 1) * 16;          // 0 or 16
    for (int kc = 0; kc < DIM; kc += BK) {
        __syncthreads();
        if (r2 < KSEL) {
            const float* src = dictn + (size_t)idxsh[r2] * DIM + kc + h16;
            #pragma unroll
            for (int j = 0; j < 16; ++j) Sc[r2][h16 + j] = src[j];
        } else {
            #pragma unroll
            for (int j = 0; j < 16; ++j) Sc[r2][h16 + j] = 0.0f;
        }
        __syncthreads();
        #pragma unroll
        for (int kk = 0; kk < BK; kk += 4) {
            v2f a;
            a[0] = Sc[wave * 16 + l][kk + 2 * half];
            a[1] = Sc[wave * 16 + l][kk + 2 * half + 1];
            #pragma unroll
            for (int nt = 0; nt < 4; ++nt) {
                v2f b;   // B[k][n] = S[n][k] -> same tile, same access pattern
                b[0] = Sc[nt * 16 + l][kk + 2 * half];
                b[1] = Sc[nt * 16 + l][kk + 2 * half + 1];
                acc[nt] = wmma_f32_16x16x4(a, b, acc[nt]);
            }
        }
    }
    __syncthreads();
    #pragma unroll
    for (int nt = 0; nt < 4; ++nt)
        #pragma unroll
        for (int r = 0; r < 8; ++r)
            G[wave * 16 + (half ? r + 8 : r)][nt * 16 + l] = acc[nt][r];
    if (tid < KPAD) vsh[tid] = (tid < KSEL) ? 1.0f : 0.0f;
    __syncthreads();

    // ---- power iteration: lambda_max of PSD Gram ----
    for (int pit = 0; pit < POWER_ITERS; ++pit) {
        if (tid < KPAD) {
            float s = 0.0f;
            #pragma unroll 8
            for (int j = 0; j < KPAD; ++j) s += G[tid][j] * vsh[j];
            tsh[tid] = s;
        }
        __syncthreads();
        red[tid] = (tid < KPAD) ? tsh[tid] * tsh[tid] : 0.0f;
        __syncthreads();
        for (int s = 64; s > 0; s >>= 1) {
            if (tid < s) red[tid] += red[tid + s];
            __syncthreads();
        }
        const float nrm = sqrtf(fmaxf(red[0], 1e-30f));
        if (tid < KPAD) vsh[tid] = tsh[tid] / nrm;
        __syncthreads();
    }
    if (tid < KPAD) {                         // Rayleigh quotient
        float s = 0.0f;
        #pragma unroll 8
        for (int j = 0; j < KPAD; ++j) s += G[tid][j] * vsh[j];
        tsh[tid] = s;
    }
    __syncthreads();
    red[tid] = (tid < KPAD) ? vsh[tid] * tsh[tid] : 0.0f;
    __syncthreads();
    for (int s = 64; s > 0; s >>= 1) {
        if (tid < s) red[tid] += red[tid + s];
        __syncthreads();
    }
    if (tid == 0) etash = 1.0f / fmaxf(red[0], 1e-20f);
    if (tid < KPAD) xsh[tid] = bsh[tid];      // x0 = b
    __syncthreads();
    const float eta = etash;

    // ---- ISTA: x = clip(x + eta*(b - G x), 0, b), 100 iterations ----
    for (int it = 0; it < ISTA_ITERS; ++it) {
        float xn = 0.0f;
        if (tid < KPAD) {
            float g = bsh[tid];
            #pragma unroll 8
            for (int j = 0; j < KPAD; ++j) g -= G[tid][j] * xsh[j];
            xn = fminf(fmaxf(xsh[tid] + eta * g, 0.0f), bsh[tid]);
        }
        __syncthreads();
        if (tid < KPAD) xsh[tid] = xn;
        __syncthreads();
    }

    // ---- L1 normalize ----
    red[tid] = (tid < KPAD) ? fabsf(xsh[tid]) : 0.0f;
    __syncthreads();
    for (int s = 64; s > 0; s >>= 1) {
        if (tid < s) red[tid] += red[tid + s];
        __syncthreads();
    }
    const float invl1 = 1.0f / fmaxf(red[0], 1e-12f);
    if (tid < KPAD) xsh[tid] *= invl1;
    __syncthreads();

    // ---- interpolated[row] = sum_k x_k * dictionary_atoms[idx_k] ----
    const int c0 = tid * 4;                   // 128 threads x 4 cols = 512
    float o0 = 0.0f, o1 = 0.0f, o2 = 0.0f, o3 = 0.0f;
    for (int k = 0; k < KSEL; ++k) {
        const float w = xsh[k];
        const float4 p = *(const float4*)(atoms + (size_t)idxsh[k] * DIM + c0);
        o0 += w * p.x; o1 += w * p.y; o2 += w * p.z; o3 += w * p.w;
    }
    float* op = out + (size_t)row * DIM + c0;
    op[0] = o0; op[1] = o1; op[2] = o2; op[3] = o3;
}

// ---------------------------------------------------------------------------
extern "C" void kernel_launch(void* const* d_in, const int* in_sizes, int n_in,
                              void* d_out, int out_size, void* d_ws, size_t ws_size,
                              hipStream_t stream) {
    (void)in_sizes; (void)n_in; (void)out_size; (void)ws_size;
    const float* batch = (const float*)d_in[0];   // (4096, 512)
    const float* atoms = (const float*)d_in[1];   // (8192, 512)
    float* out = (float*)d_out;                   // (4096, 512)

    // Workspace layout (~154 MB): dict_norm | sims | sub_sims | sub_idx
    float* dictn    = (float*)d_ws;
    float* sims     = dictn + (size_t)ATOMS * DIM;
    float* sub_sims = sims + (size_t)BATCH * ATOMS;
    int*   sub_idx  = (int*)(sub_sims + (size_t)BATCH * KPAD);

    k_normalize_dict<<<ATOMS, 128, 0, stream>>>(atoms, dictn);
    k_sims_gemm<<<dim3(ATOMS / BN, BATCH / BM), 128, 0, stream>>>(batch, dictn, sims);
    k_topk<<<BATCH, 256, 0, stream>>>(sims, sub_sims, sub_idx);
    k_nnk_solve<<<BATCH, 128, 0, stream>>>(dictn, atoms, sub_sims, sub_idx, out);
}